// dnn_crf_52544629899692
// MI455X (gfx1250) — compile-verified
//
#include <hip/hip_runtime.h>

#define B_ 2048
#define T_ 80
#define L_ 128
#define ROWS 16     // batch rows per workgroup
#define PPAD 4      // pad so stride-132 A-fragment ds_load_b64 hits distinct banks
#define APAD 8      // pad so column-strided alpha stores hit distinct banks

typedef __attribute__((ext_vector_type(2))) float v2f;
typedef __attribute__((ext_vector_type(8))) float v8f;

// ---- kernel 0: E = exp(trans), one-time, stays hot in L2 ----
__global__ __launch_bounds__(256) void exp_trans_kernel(const float* __restrict__ trans,
                                                        float* __restrict__ E) {
    int i = blockIdx.x * blockDim.x + threadIdx.x;
    E[i] = __expf(trans[i]);
}

// ---- kernel 1: gold-path score (numerator) per batch row ----
__global__ __launch_bounds__(256) void numerator_kernel(const float* __restrict__ x,
                                                        const float* __restrict__ trans,
                                                        const float* __restrict__ st,
                                                        const float* __restrict__ et,
                                                        const int* __restrict__ y,
                                                        float* __restrict__ out) {
    int b = blockIdx.x * blockDim.x + threadIdx.x;
    const int* yb = y + b * T_;
    const float* xb = x + (size_t)b * T_ * L_;
    int yp = yb[0];
    float s = st[yp] + xb[yp];
    for (int t = 1; t < T_; ++t) {
        int yt = yb[t];
        s += xb[t * L_ + yt] + trans[yp * L_ + yt];
        yp = yt;
    }
    out[b] = s + et[yp];
}

// ---- kernel 2: forward algorithm via exp-domain WMMA GEMM ----
__global__ __launch_bounds__(256) void crf_forward_kernel(const float* __restrict__ x,
                                                          const float* __restrict__ E,
                                                          const float* __restrict__ st,
                                                          const float* __restrict__ et,
                                                          float* __restrict__ out) {
    __shared__ float sP[ROWS][L_ + PPAD];   // P = exp(alpha - M)
    __shared__ float sA[ROWS][L_ + APAD];   // alpha
    __shared__ float sM[ROWS];              // per-row max

    const int tid  = threadIdx.x;
    const int lane = tid & 31;
    const int wave = tid >> 5;              // 0..7, owns output cols [wave*16, wave*16+16)
    const int b0   = blockIdx.x * ROWS;

    const int n0 = wave * 16;
    const int hi = (lane >= 16) ? 1 : 0;
    const int nc = n0 + (lane & 15);        // this lane's output column

    // B fragments of the fixed matrix E (128x16 tile), held in registers for all 80 steps.
    // V_WMMA layout, B 4x16: lanes 0-15 hold {K=k0, K=k0+1}, lanes 16-31 hold {K=k0+2, K=k0+3}.
    v2f bf[32];
#pragma unroll
    for (int kk = 0; kk < 32; ++kk) {
        int k0 = kk * 4 + hi * 2;
        v2f b;
        b.x = E[k0 * L_ + nc];
        b.y = E[(k0 + 1) * L_ + nc];
        bf[kk] = b;
    }

    // alpha(t=0) = start_trans + x[:,0,:]
    {
        int r  = tid >> 4;                  // 16 threads per row
        int c0 = (tid & 15) * 8;
        const float* xr = x + (size_t)(b0 + r) * T_ * L_;
#pragma unroll
        for (int c = 0; c < 8; ++c)
            sA[r][c0 + c] = st[c0 + c] + xr[c0 + c];
    }
    __syncthreads();

    const int rr = wave * 2 + hi;           // row handled by this lane in reduce phase
    const int cc = (lane & 15) * 8;

    for (int t = 1; t < T_; ++t) {
        // --- step 1: row max + P = exp(alpha - max) ---
        float av[8];
#pragma unroll
        for (int c = 0; c < 8; ++c) av[c] = sA[rr][cc + c];
        float m = av[0];
#pragma unroll
        for (int c = 1; c < 8; ++c) m = fmaxf(m, av[c]);
#pragma unroll
        for (int off = 1; off < 16; off <<= 1)
            m = fmaxf(m, __shfl_xor(m, off, 16));
        if ((lane & 15) == 0) sM[rr] = m;
#pragma unroll
        for (int c = 0; c < 8; ++c) sP[rr][cc + c] = __expf(av[c] - m);
        __syncthreads();

        // --- step 2: S = P * E, 16x16 tile per wave, K=128 as 32 fp32 WMMAs ---
        v8f acc = {};
#pragma unroll
        for (int kk = 0; kk < 32; ++kk) {
            int k0 = kk * 4 + hi * 2;
            int ar = lane & 15;             // A 16x4: lanes 0-15 rows, lanes 16-31 same rows K+2
            v2f a = *(const v2f*)&sP[ar][k0];
            acc = __builtin_amdgcn_wmma_f32_16x16x4_f32(false, a, false, bf[kk],
                                                        (short)0, acc, false, false);
        }

        // --- step 3: alpha_new = log(S) + M + x[:,t,:] ---
#pragma unroll
        for (int v = 0; v < 8; ++v) {
            int r = v + hi * 8;             // C/D layout: VGPR v = row v (+8 upper half)
            const float* xr = x + ((size_t)(b0 + r) * T_ + t) * L_;
            sA[r][nc] = __logf(acc[v]) + sM[r] + xr[nc];
            if (t + 1 < T_) __builtin_prefetch(xr + L_ + nc, 0, 1);
        }
        __syncthreads();
    }

    // --- final: log_z = logsumexp(alpha + end_trans); out = numerator - log_z ---
    float av[8];
#pragma unroll
    for (int c = 0; c < 8; ++c) av[c] = sA[rr][cc + c] + et[cc + c];
    float m = av[0];
#pragma unroll
    for (int c = 1; c < 8; ++c) m = fmaxf(m, av[c]);
#pragma unroll
    for (int off = 1; off < 16; off <<= 1)
        m = fmaxf(m, __shfl_xor(m, off, 16));
    float s = 0.f;
#pragma unroll
    for (int c = 0; c < 8; ++c) s += __expf(av[c] - m);
#pragma unroll
    for (int off = 1; off < 16; off <<= 1)
        s += __shfl_xor(s, off, 16);
    if ((lane & 15) == 0) {
        out[b0 + rr] -= m + __logf(s);
    }
}

extern "C" void kernel_launch(void* const* d_in, const int* in_sizes, int n_in,
                              void* d_out, int out_size, void* d_ws, size_t ws_size,
                              hipStream_t stream) {
    const float* x  = (const float*)d_in[0];   // [B,T,L]
    const float* tr = (const float*)d_in[1];   // [L,L]
    const float* st = (const float*)d_in[2];   // [L]
    const float* et = (const float*)d_in[3];   // [L]
    const int*   y  = (const int*)d_in[4];     // [B,T]
    float* out = (float*)d_out;                // [B]
    float* E   = (float*)d_ws;                 // [L,L] = 64 KB scratch

    exp_trans_kernel<<<(L_ * L_) / 256, 256, 0, stream>>>(tr, E);
    numerator_kernel<<<B_ / 256, 256, 0, stream>>>(x, tr, st, et, y, out);
    crf_forward_kernel<<<B_ / ROWS, 256, 0, stream>>>(x, E, st, et, out);
}